// GCN_54752243089440
// MI455X (gfx1250) — compile-verified
//
#include <hip/hip_runtime.h>

#define NUM_NODES   65536
#define GCN_EMB_DIM 128
#define N_EMBD      512
#define NUM_EDGES   1048576
#define NEG_SLOPE   0.1f
#define NUM_GRAPHS  256
#define GEMM_K      32768   // 128 * 256

typedef float v2f __attribute__((ext_vector_type(2)));
typedef float v8f __attribute__((ext_vector_type(8)));

#if defined(__HIP_DEVICE_COMPILE__) && __has_builtin(__builtin_amdgcn_wmma_f32_16x16x4_f32)
#define HAVE_WMMA_F32X4 1
#else
#define HAVE_WMMA_F32X4 0
#endif

// Async global->LDS copy (CDNA5 GLOBAL_LOAD_ASYNC_TO_LDS_B128, tracked by ASYNCcnt).
// The LDS destination address is the low 32 bits of the generic shared pointer
// (flat LDS aperture).
__device__ __forceinline__ void async_g2l_b128(void* lds, const void* gptr) {
    unsigned lds_off = (unsigned)(size_t)lds;
    asm volatile("global_load_async_to_lds_b128 %0, %1, off"
                 :: "v"(lds_off), "v"(gptr)
                 : "memory");
}
__device__ __forceinline__ void async_wait0() {
    asm volatile("s_wait_asynccnt 0x0" ::: "memory");
}

// ---------------- degree / normalization ----------------
__global__ void k_deg_init(float* __restrict__ deg) {
    int i = blockIdx.x * blockDim.x + threadIdx.x;
    if (i < NUM_NODES) deg[i] = 1.0f;              // self-loop
}

__global__ void k_deg_accum(const int* __restrict__ dst, float* __restrict__ deg) {
    int e = blockIdx.x * blockDim.x + threadIdx.x;
    if (e < NUM_EDGES) atomicAdd(&deg[dst[e]], 1.0f);
}

__global__ void k_dinv(float* __restrict__ deg) {
    int i = blockIdx.x * blockDim.x + threadIdx.x;
    if (i < NUM_NODES) deg[i] = rsqrtf(deg[i]);    // deg >= 1 always (self-loop)
}

// ---------------- GCNConv 1: 3 -> 3 ----------------
// h1s[i] = (x[i] @ W1) * dinv[i]           (messages, pre-scaled by source norm)
// out1[i] = (x[i] @ W1) * dinv[i]^2 + b1   (self-loop contribution + bias seed)
__global__ void k_conv1_node(const float* __restrict__ x,
                             const float* __restrict__ W1, const float* __restrict__ b1,
                             const float* __restrict__ dinv,
                             float* __restrict__ h1s, float* __restrict__ out1) {
    int i = blockIdx.x * blockDim.x + threadIdx.x;
    if (i >= NUM_NODES) return;
    float x0 = x[i * 3 + 0], x1 = x[i * 3 + 1], x2 = x[i * 3 + 2];
    float di = dinv[i];
#pragma unroll
    for (int j = 0; j < 3; ++j) {
        float h = x0 * W1[0 * 3 + j] + x1 * W1[1 * 3 + j] + x2 * W1[2 * 3 + j];
        h1s[i * 3 + j]  = h * di;
        out1[i * 3 + j] = h * di * di + b1[j];
    }
}

__global__ void k_conv1_edge(const int* __restrict__ src, const int* __restrict__ dst,
                             const float* __restrict__ dinv,
                             const float* __restrict__ h1s, float* __restrict__ out1) {
    int e = blockIdx.x * blockDim.x + threadIdx.x;
    if (e >= NUM_EDGES) return;
    int s = src[e], d = dst[e];
    float nd = dinv[d];
    atomicAdd(&out1[d * 3 + 0], h1s[s * 3 + 0] * nd);
    atomicAdd(&out1[d * 3 + 1], h1s[s * 3 + 1] * nd);
    atomicAdd(&out1[d * 3 + 2], h1s[s * 3 + 2] * nd);
}

// ---------------- GCNConv 2: 3 -> 128 (leaky_relu fused on input) ----------------
__global__ void k_conv2_node(const float* __restrict__ out1,
                             const float* __restrict__ W2, const float* __restrict__ b2,
                             const float* __restrict__ dinv,
                             float* __restrict__ h2s, float* __restrict__ out2) {
    int idx = blockIdx.x * blockDim.x + threadIdx.x;   // NUM_NODES * 128 threads
    int node = idx >> 7;
    int j = idx & 127;
    float a0 = out1[node * 3 + 0]; a0 = a0 >= 0.0f ? a0 : NEG_SLOPE * a0;
    float a1 = out1[node * 3 + 1]; a1 = a1 >= 0.0f ? a1 : NEG_SLOPE * a1;
    float a2 = out1[node * 3 + 2]; a2 = a2 >= 0.0f ? a2 : NEG_SLOPE * a2;
    float h = a0 * W2[0 * 128 + j] + a1 * W2[1 * 128 + j] + a2 * W2[2 * 128 + j];
    float di = dinv[node];
    float hs = h * di;
    h2s[idx]  = hs;                 // pre-scaled message
    out2[idx] = hs * di + b2[j];    // self-loop + bias seed
}

// One wave32 per edge, each lane owns 4 features (float4 gather, 4 L2 atomics).
__global__ void k_conv2_edge(const int* __restrict__ src, const int* __restrict__ dst,
                             const float* __restrict__ dinv,
                             const float* __restrict__ h2s, float* __restrict__ out2) {
    int lane = threadIdx.x & 31;
    long e = (long)blockIdx.x * 8 + (threadIdx.x >> 5);   // blockDim = 256 = 8 waves
    if (e >= NUM_EDGES) return;
    int s = src[e], d = dst[e];
    float nd = dinv[d];
    const float4 v = *(const float4*)(h2s + (size_t)s * 128 + lane * 4);
    float* o = out2 + (size_t)d * 128 + lane * 4;
    atomicAdd(o + 0, v.x * nd);
    atomicAdd(o + 1, v.y * nd);
    atomicAdd(o + 2, v.z * nd);
    atomicAdd(o + 3, v.w * nd);
}

// ---------------- Final GEMM: C[256,512] = A[256,32768] @ B[32768,512] + b3 ----------------
// Block: 256 threads = 8 waves (wave32). Block tile 32x64, wave tile 16x16, K-panel 32.
// K-panels are double-buffered in LDS via GLOBAL_LOAD_ASYNC_TO_LDS_B128 so the
// async DMA of panel i+1 overlaps the V_WMMA_F32_16X16X4_F32 work on panel i.
// Fragment layouts per CDNA5 ISA 7.12.2:
//   A 16x4 f32: lanes 0-15 hold M=lane, VGPR{0,1}=K{0,1}; lanes 16-31 K{2,3}.
//   B 4x16 f32: VGPR0 rows K0 (lanes 0-15) / K2 (lanes 16-31); VGPR1 rows K1/K3.
//   C/D 16x16:  VGPR i -> M=i (lanes 0-15), M=i+8 (lanes 16-31), N=lane&15.
__launch_bounds__(256)
__global__ void k_gemm(const float* __restrict__ A, const float* __restrict__ B,
                       const float* __restrict__ bias, float* __restrict__ C) {
    __shared__ float As[2][32][36];   // +4 pad: rows 144 B (16B aligned, spreads banks)
    __shared__ float Bs[2][32][68];   // +4 pad: rows 272 B

    const int K = GEMM_K, N = N_EMBD;
    const int m0 = blockIdx.x * 32;
    const int n0 = blockIdx.y * 64;
    const int t = threadIdx.x;
    const int lane = t & 31;
    const int wave = t >> 5;
    const int m_off = (wave >> 2) * 16;   // 0 / 16
    const int n_off = (wave & 3) * 16;    // 0 / 16 / 32 / 48
    const int hf  = lane >> 4;            // half-wave id
    const int l16 = lane & 15;

    // cooperative panel-load indices (one b128 async copy per source line)
    const int arow = t >> 3, akq = (t & 7) << 2;     // A: 32 rows x 32 k
    const int brow = t >> 4, bcq = (t & 15) << 2;    // B: 32 k x 64 n (two copies/thread)

    const float* gA  = A + (size_t)(m0 + arow) * K + akq;
    const float* gB0 = B + (size_t)brow * N + n0 + bcq;
    const float* gB1 = B + (size_t)(brow + 16) * N + n0 + bcq;

    // prologue: stage panel 0 into buffer 0
    async_g2l_b128(&As[0][arow][akq],      gA);
    async_g2l_b128(&Bs[0][brow][bcq],      gB0);
    async_g2l_b128(&Bs[0][brow + 16][bcq], gB1);

    v8f acc = {0.f, 0.f, 0.f, 0.f, 0.f, 0.f, 0.f, 0.f};

    const int NITER = K / 32;             // 1024
    for (int i = 0; i < NITER; ++i) {
        const int buf = i & 1;
        async_wait0();                    // this wave's async copies for `buf` done
        __syncthreads();                  // all waves' copies visible; prior reads done

        if (i + 1 < NITER) {              // overlap next panel's DMA with compute
            const int nb = buf ^ 1;
            const size_t kk = (size_t)(i + 1) * 32;
            async_g2l_b128(&As[nb][arow][akq],      gA + kk);
            async_g2l_b128(&Bs[nb][brow][bcq],      gB0 + kk * N);
            async_g2l_b128(&Bs[nb][brow + 16][bcq], gB1 + kk * N);
        }

#if HAVE_WMMA_F32X4
#pragma unroll
        for (int k4 = 0; k4 < 32; k4 += 4) {
            v2f a, b;
            a.x = As[buf][m_off + l16][k4 + hf * 2 + 0];
            a.y = As[buf][m_off + l16][k4 + hf * 2 + 1];
            b.x = Bs[buf][k4 + hf * 2 + 0][n_off + l16];
            b.y = Bs[buf][k4 + hf * 2 + 1][n_off + l16];
            acc = __builtin_amdgcn_wmma_f32_16x16x4_f32(
                /*neg_a=*/false, a, /*neg_b=*/false, b,
                /*c_mod=*/(short)0, acc, /*reuse_a=*/false, /*reuse_b=*/false);
        }
#else
        // scalar fallback with the identical output-lane mapping
#pragma unroll
        for (int k = 0; k < 32; ++k) {
            float bv = Bs[buf][k][n_off + l16];
#pragma unroll
            for (int ii = 0; ii < 8; ++ii)
                acc[ii] += As[buf][m_off + ii + hf * 8][k] * bv;
        }
#endif
        __syncthreads();                  // reads of `buf` done before it is re-filled
    }

    const int col = n0 + n_off + l16;
    const float bb = bias[col];
#pragma unroll
    for (int i = 0; i < 8; ++i) {
        int row = m0 + m_off + i + hf * 8;
        C[(size_t)row * N + col] = acc[i] + bb;
    }
}

// ---------------- launch ----------------
extern "C" void kernel_launch(void* const* d_in, const int* in_sizes, int n_in,
                              void* d_out, int out_size, void* d_ws, size_t ws_size,
                              hipStream_t stream) {
    (void)in_sizes; (void)n_in; (void)out_size; (void)ws_size;

    const float* nodes = (const float*)d_in[0];
    const int*   ei    = (const int*)d_in[1];
    const float* W1    = (const float*)d_in[2];
    const float* b1    = (const float*)d_in[3];
    const float* W2    = (const float*)d_in[4];
    const float* b2    = (const float*)d_in[5];
    const float* W3    = (const float*)d_in[6];
    const float* b3    = (const float*)d_in[7];
    float* out = (float*)d_out;

    // workspace layout (floats): dinv | h1s | out1 | h2s | out2  (~66 MB)
    float* ws   = (float*)d_ws;
    float* dinv = ws;
    float* h1s  = dinv + NUM_NODES;
    float* out1 = h1s + (size_t)NUM_NODES * 3;
    float* h2s  = out1 + (size_t)NUM_NODES * 3;
    float* out2 = h2s + (size_t)NUM_NODES * GCN_EMB_DIM;

    const int* src = ei;               // edge_index[0]
    const int* dst = ei + NUM_EDGES;   // edge_index[1]

    k_deg_init <<<NUM_NODES / 256, 256, 0, stream>>>(dinv);
    k_deg_accum<<<NUM_EDGES / 256, 256, 0, stream>>>(dst, dinv);
    k_dinv     <<<NUM_NODES / 256, 256, 0, stream>>>(dinv);

    k_conv1_node<<<NUM_NODES / 256, 256, 0, stream>>>(nodes, W1, b1, dinv, h1s, out1);
    k_conv1_edge<<<NUM_EDGES / 256, 256, 0, stream>>>(src, dst, dinv, h1s, out1);

    k_conv2_node<<<(NUM_NODES * GCN_EMB_DIM) / 256, 256, 0, stream>>>(out1, W2, b2, dinv, h2s, out2);
    k_conv2_edge<<<NUM_EDGES / 8, 256, 0, stream>>>(src, dst, dinv, h2s, out2);

    k_gemm<<<dim3(NUM_GRAPHS / 32, N_EMBD / 64), 256, 0, stream>>>(out2, W3, b3, out);
}